// TRBlock_41231686042367
// MI455X (gfx1250) — compile-verified
//
#include <hip/hip_runtime.h>
#include <hip/hip_bf16.h>
#include <cstdint>

#define N_PTS   100000
#define P_CH    96
#define VEC_CH  4
#define M_EDGES 70000
#define K_MAPS  43
#define CONV_MAPS 27

typedef __attribute__((ext_vector_type(2))) float v2f;
typedef __attribute__((ext_vector_type(8))) float v8f;

__device__ __forceinline__ float wave_reduce_add(float v) {
#pragma unroll
    for (int off = 16; off > 0; off >>= 1) v += __shfl_xor(v, off, 32);
    return v;
}

// ---------------------------------------------------------------------------
// Sparse conv, 96->96, one kernel map per blockIdx.y.
// Each wave: 16 gathered edge-rows x 96 output cols via V_WMMA_F32_16X16X4_F32.
// Weights staged transposed in LDS so B fragments are single ds_load_b64.
// ---------------------------------------------------------------------------
__global__ __launch_bounds__(256) void spconv96x96_wmma(
    const float* __restrict__ x, const float* __restrict__ W,
    const int* __restrict__ kmaps, float* __restrict__ out)
{
    __shared__ float WldsT[96 * 96];           // [col][row] (transposed)
    const int map = blockIdx.y;
    const int tid = threadIdx.x;

    const float* Wm = W + (size_t)map * 96 * 96;
    for (int j = tid; j < 96 * 96; j += 256) {
        int r = j / 96, c = j % 96;
        WldsT[c * 96 + r] = Wm[j];
    }
    __syncthreads();

    const int wave = tid >> 5;
    const int lane = tid & 31;
    const int lo   = lane & 15;
    const int hi   = lane >> 4;                // 0 or 1
    const int numTiles = M_EDGES / 16;         // 4375 exactly
    const int tile = blockIdx.x * 8 + wave;
    if (tile >= numTiles) return;              // wave-uniform: EXEC stays all-ones

    const int  ebase = tile * 16;
    const int* km    = kmaps + (size_t)map * M_EDGES * 2;
    const int  srcRow = km[(ebase + lo) * 2 + 0];
    const int  dstRow = km[(ebase + lo) * 2 + 1];

    v8f acc[6] = {};                           // 6 col tiles of 16
    const float* arow = x + (size_t)srcRow * 96 + 2 * hi;

    for (int k0 = 0; k0 < 96; k0 += 4) {
        // A fragment: lane l<16 -> (M=l, K=k0..k0+1); l>=16 -> (M=l-16, K=k0+2..k0+3)
        v2f av = *(const v2f*)(arow + k0);
#pragma unroll
        for (int ct = 0; ct < 6; ++ct) {
            int col = ct * 16 + lo;
            v2f bv = *(const v2f*)(&WldsT[col * 96 + k0 + 2 * hi]);
            acc[ct] = __builtin_amdgcn_wmma_f32_16x16x4_f32(
                false, av, false, bv, (short)0, acc[ct], false, false);
        }
    }

    // C layout: vgpr r -> (M = r + hi*8, N = lo). Scatter-add to out[dst].
#pragma unroll
    for (int ct = 0; ct < 6; ++ct) {
#pragma unroll
        for (int r = 0; r < 8; ++r) {
            int m    = r + hi * 8;
            int drow = __shfl(dstRow, m, 32);  // lane m holds dst of edge-row m
            atomicAdd(out + (size_t)drow * 96 + ct * 16 + lo, acc[ct][r]);
        }
    }
}

// ---------------------------------------------------------------------------
// Sparse conv 96->4 (Wq2). Wave per edge, shuffle-reduced dot products.
// ---------------------------------------------------------------------------
__global__ __launch_bounds__(256) void spconv96x4(
    const float* __restrict__ xin, const float* __restrict__ W,
    const int* __restrict__ kmaps, float* __restrict__ out4)
{
    const int map  = blockIdx.y;
    const int wave = threadIdx.x >> 5;
    const int lane = threadIdx.x & 31;
    const int e    = blockIdx.x * 8 + wave;
    if (e >= M_EDGES) return;
    const int* km = kmaps + (size_t)map * M_EDGES * 2;
    const int  s = km[e * 2 + 0], d = km[e * 2 + 1];
    const float* row = xin + (size_t)s * 96;
    const float* Wm  = W + (size_t)map * 96 * 4;

    float p0 = 0, p1 = 0, p2 = 0, p3 = 0;
#pragma unroll
    for (int t = 0; t < 3; ++t) {
        int c = lane + t * 32;
        float q = row[c];
        const float* w = Wm + c * 4;
        p0 += q * w[0]; p1 += q * w[1]; p2 += q * w[2]; p3 += q * w[3];
    }
    p0 = wave_reduce_add(p0); p1 = wave_reduce_add(p1);
    p2 = wave_reduce_add(p2); p3 = wave_reduce_add(p3);
    if (lane == 0) {
        atomicAdd(out4 + (size_t)d * 4 + 0, p0);
        atomicAdd(out4 + (size_t)d * 4 + 1, p1);
        atomicAdd(out4 + (size_t)d * 4 + 2, p2);
        atomicAdd(out4 + (size_t)d * 4 + 3, p3);
    }
}

// ---------------------------------------------------------------------------
// Per-column sum / sumsq for BatchNorm (biased var). LDS atomics -> global.
// ---------------------------------------------------------------------------
__global__ __launch_bounds__(256) void colstats(
    const float* __restrict__ xin, float* __restrict__ stats,
    long long total, int C)
{
    __shared__ float ls[192];
    for (int j = threadIdx.x; j < 2 * C; j += 256) ls[j] = 0.f;
    __syncthreads();
    long long stride = (long long)gridDim.x * 256;
    for (long long idx = (long long)blockIdx.x * 256 + threadIdx.x; idx < total; idx += stride) {
        float v = xin[idx];
        int c = (int)(idx % C);
        atomicAdd(&ls[c], v);
        atomicAdd(&ls[C + c], v * v);
    }
    __syncthreads();
    for (int j = threadIdx.x; j < 2 * C; j += 256) atomicAdd(&stats[j], ls[j]);
}

// ---------------------------------------------------------------------------
// y = relu(g*(x-m)*rsqrt(v+eps)+b) [+ repeat(pe)] [+ residual x]
// ---------------------------------------------------------------------------
__global__ void bn_apply(
    const float* __restrict__ src, float* __restrict__ dst,
    const float* __restrict__ stats, const float* __restrict__ g,
    const float* __restrict__ bb, const float* __restrict__ pe,
    const float* __restrict__ xres, int n, int C)
{
    long long idx   = (long long)blockIdx.x * blockDim.x + threadIdx.x;
    long long total = (long long)n * C;
    if (idx >= total) return;
    int c = (int)(idx % C);
    long long row = idx / C;
    float invn = 1.0f / (float)n;
    float mean = stats[c] * invn;
    float var  = stats[C + c] * invn - mean * mean;
    float y = g[c] * (src[idx] - mean) * rsqrtf(var + 1e-5f) + bb[c];
    y = fmaxf(y, 0.f);
    if (pe)   y += pe[row * 4 + c / 24];       // jnp.repeat(pe, 24, axis=1)
    if (xres) y += xres[idx];
    dst[idx] = y;
}

// ---------------------------------------------------------------------------
// Positional path: coords(N,3) @ Wp1(3,96), then h1(N,96) @ Wp2(96,4)
// ---------------------------------------------------------------------------
__global__ void pe_h1(const float* __restrict__ coords, const float* __restrict__ Wp1,
                      float* __restrict__ h1, int n)
{
    long long idx = (long long)blockIdx.x * blockDim.x + threadIdx.x;
    if (idx >= (long long)n * 96) return;
    long long i = idx / 96; int c = (int)(idx % 96);
    h1[idx] = coords[i * 3 + 0] * Wp1[c]
            + coords[i * 3 + 1] * Wp1[96 + c]
            + coords[i * 3 + 2] * Wp1[192 + c];
}

__global__ void pe_h2(const float* __restrict__ h1, const float* __restrict__ Wp2,
                      float* __restrict__ pe4, int n)
{
    int i = blockIdx.x * blockDim.x + threadIdx.x;
    if (i >= n) return;
    float a0 = 0, a1 = 0, a2 = 0, a3 = 0;
    const float* r = h1 + (size_t)i * 96;
    for (int c = 0; c < 96; ++c) {
        float v = r[c];
        const float* w = Wp2 + c * 4;
        a0 += v * w[0]; a1 += v * w[1]; a2 += v * w[2]; a3 += v * w[3];
    }
    pe4[i * 4 + 0] = a0; pe4[i * 4 + 1] = a1;
    pe4[i * 4 + 2] = a2; pe4[i * 4 + 3] = a3;
}

// ---------------------------------------------------------------------------
// Inverse maps: inv[k][dst] = src (dst is unique within a map). inv pre-set -1.
// ---------------------------------------------------------------------------
__global__ void build_inv(const int* __restrict__ kmaps, int* __restrict__ inv)
{
    long long idx = (long long)blockIdx.x * blockDim.x + threadIdx.x;
    if (idx >= (long long)K_MAPS * M_EDGES) return;
    int k = (int)(idx / M_EDGES), e = (int)(idx % M_EDGES);
    const int* p = kmaps + ((long long)k * M_EDGES + e) * 2;
    inv[(long long)k * N_PTS + p[1]] = p[0];
}

// ---------------------------------------------------------------------------
// Attention logits: wave per (map k, row i). d = qf4[s]-qf4[i];
// logits = (relu(d@W1+b1)@W2+b2)*mask, mask = (s>=0 && sum|x[s]|>0)
// ---------------------------------------------------------------------------
__global__ __launch_bounds__(256) void attn_logits(
    const float* __restrict__ x, const float* __restrict__ qf4,
    const int* __restrict__ inv,
    const float* __restrict__ W1, const float* __restrict__ b1,
    const float* __restrict__ W2, const float* __restrict__ b2,
    float* __restrict__ logits, float* __restrict__ masks)
{
    const int wave = threadIdx.x >> 5, lane = threadIdx.x & 31;
    long long gw = (long long)blockIdx.x * 8 + wave;
    if (gw >= (long long)K_MAPS * N_PTS) return;
    int k = (int)(gw / N_PTS);
    int i = (int)(gw % N_PTS);
    int s = inv[(long long)k * N_PTS + i];

    float sumabs = 0, p0 = 0, p1 = 0, p2 = 0, p3 = 0;
    if (s >= 0) {  // wave-uniform
        float d0 = qf4[(size_t)s * 4 + 0] - qf4[(size_t)i * 4 + 0];
        float d1 = qf4[(size_t)s * 4 + 1] - qf4[(size_t)i * 4 + 1];
        float d2 = qf4[(size_t)s * 4 + 2] - qf4[(size_t)i * 4 + 2];
        float d3 = qf4[(size_t)s * 4 + 3] - qf4[(size_t)i * 4 + 3];
        const float* xr = x + (size_t)s * 96;
#pragma unroll
        for (int t = 0; t < 3; ++t) {
            int j = lane + t * 32;
            sumabs += fabsf(xr[j]);
            float h = b1[j] + d0 * W1[j] + d1 * W1[96 + j]
                            + d2 * W1[192 + j] + d3 * W1[288 + j];
            h = fmaxf(h, 0.f);
            p0 += h * W2[j * 4 + 0]; p1 += h * W2[j * 4 + 1];
            p2 += h * W2[j * 4 + 2]; p3 += h * W2[j * 4 + 3];
        }
    }
    sumabs = wave_reduce_add(sumabs);
    p0 = wave_reduce_add(p0); p1 = wave_reduce_add(p1);
    p2 = wave_reduce_add(p2); p3 = wave_reduce_add(p3);
    if (lane == 0) {
        float mk = (s >= 0 && sumabs > 0.f) ? 1.f : 0.f;
        float* lp = logits + ((long long)k * N_PTS + i) * 4;
        lp[0] = (p0 + b2[0]) * mk; lp[1] = (p1 + b2[1]) * mk;
        lp[2] = (p2 + b2[2]) * mk; lp[3] = (p3 + b2[3]) * mk;
        masks[(long long)k * N_PTS + i] = mk;
    }
}

// Softmax over the 43-map axis, in place; thread per (row, vec-channel).
__global__ void attn_softmax(float* __restrict__ logits)
{
    long long idx = (long long)blockIdx.x * blockDim.x + threadIdx.x;
    if (idx >= (long long)N_PTS * 4) return;
    float v[K_MAPS];
    float mx = -3.4e38f;
    for (int k = 0; k < K_MAPS; ++k) {
        v[k] = logits[(long long)k * N_PTS * 4 + idx];
        mx = fmaxf(mx, v[k]);
    }
    float sum = 0.f;
    for (int k = 0; k < K_MAPS; ++k) { v[k] = __expf(v[k] - mx); sum += v[k]; }
    float is = 1.f / sum;
    for (int k = 0; k < K_MAPS; ++k) logits[(long long)k * N_PTS * 4 + idx] = v[k] * is;
}

// Output accumulation as a gather: wave per row i, lane covers 3 channels.
__global__ __launch_bounds__(256) void out_accum(
    const float* __restrict__ v_f, const float* __restrict__ attn,
    const float* __restrict__ masks, const int* __restrict__ inv,
    float* __restrict__ outpre)
{
    int wave = threadIdx.x >> 5, lane = threadIdx.x & 31;
    int i = blockIdx.x * 8 + wave;
    if (i >= N_PTS) return;
    float a0 = 0, a1 = 0, a2 = 0;
    for (int k = 0; k < K_MAPS; ++k) {
        int s = inv[(long long)k * N_PTS + i];
        if (s < 0) continue;                      // wave-uniform
        float mk = masks[(long long)k * N_PTS + i];
        if (mk == 0.f) continue;
        int ai = (k < 27) ? k : (k < 35) ? (k - 27) : (k - 35);   // ATTN_IDX
        const float* ap = attn + ((long long)ai * N_PTS + i) * 4;
        const float* vr = v_f + (size_t)s * 96;
        a0 += vr[lane]      * ap[lane / 24]        * mk;
        a1 += vr[lane + 32] * ap[(lane + 32) / 24] * mk;
        a2 += vr[lane + 64] * ap[(lane + 64) / 24] * mk;
    }
    float* orow = outpre + (size_t)i * 96;
    orow[lane] = a0; orow[lane + 32] = a1; orow[lane + 64] = a2;
}

// ---------------------------------------------------------------------------
extern "C" void kernel_launch(void* const* d_in, const int* in_sizes, int n_in,
                              void* d_out, int out_size, void* d_ws, size_t ws_size,
                              hipStream_t stream)
{
    const float* x     = (const float*)d_in[0];
    const float* coords= (const float*)d_in[1];
    const float* Wq1   = (const float*)d_in[2];
    const float* gq1   = (const float*)d_in[3];
    const float* bq1   = (const float*)d_in[4];
    const float* Wq2   = (const float*)d_in[5];
    const float* gq2   = (const float*)d_in[6];
    const float* bq2   = (const float*)d_in[7];
    const float* Wv    = (const float*)d_in[8];
    const float* gv    = (const float*)d_in[9];
    const float* bv    = (const float*)d_in[10];
    const float* Wp1   = (const float*)d_in[11];
    const float* gp1   = (const float*)d_in[12];
    const float* bp1   = (const float*)d_in[13];
    const float* Wp2   = (const float*)d_in[14];
    const float* gp2   = (const float*)d_in[15];
    const float* bp2   = (const float*)d_in[16];
    const float* W1    = (const float*)d_in[17];
    const float* b1    = (const float*)d_in[18];
    const float* W2    = (const float*)d_in[19];
    const float* b2    = (const float*)d_in[20];
    const float* g_out = (const float*)d_in[21];
    const float* b_out = (const float*)d_in[22];
    const int*   kmaps = (const int*)d_in[23];

    char* ws = (char*)d_ws;
    size_t off = 0;
    auto alloc = [&](size_t bytes) -> void* {
        off = (off + 255) & ~(size_t)255;
        void* p = ws + off; off += bytes; return p;
    };
    float* qf1    = (float*)alloc((size_t)N_PTS * 96 * 4);
    float* vf     = (float*)alloc((size_t)N_PTS * 96 * 4);
    float* h1     = (float*)alloc((size_t)N_PTS * 96 * 4);
    float* outp   = (float*)alloc((size_t)N_PTS * 96 * 4);
    float* qf4    = (float*)alloc((size_t)N_PTS * 4 * 4);
    float* pe4    = (float*)alloc((size_t)N_PTS * 4 * 4);
    float* logits = (float*)alloc((size_t)K_MAPS * N_PTS * 4 * 4);
    float* masks  = (float*)alloc((size_t)K_MAPS * N_PTS * 4);
    int*   inv    = (int*)  alloc((size_t)K_MAPS * N_PTS * 4);
    float* stats  = (float*)alloc(256 * 4);
    (void)in_sizes; (void)n_in; (void)out_size; (void)ws_size;

    const int NT  = (M_EDGES / 16 + 7) / 8;            // wmma conv grid.x
    const int EB  = (M_EDGES + 7) / 8;                 // wave-per-edge grid.x
    const int EL96= (int)(((long long)N_PTS * 96 + 255) / 256);
    const int EL4 = (N_PTS * 4 + 255) / 256;

    // qf1 = relu(bn(spconv(x, Wq1)))
    hipMemsetAsync(qf1, 0, (size_t)N_PTS * 96 * 4, stream);
    spconv96x96_wmma<<<dim3(NT, CONV_MAPS), 256, 0, stream>>>(x, Wq1, kmaps, qf1);
    hipMemsetAsync(stats, 0, 1024, stream);
    colstats<<<1024, 256, 0, stream>>>(qf1, stats, (long long)N_PTS * 96, 96);
    bn_apply<<<EL96, 256, 0, stream>>>(qf1, qf1, stats, gq1, bq1, nullptr, nullptr, N_PTS, 96);

    // qf4 = relu(bn(spconv4(qf1, Wq2)))
    hipMemsetAsync(qf4, 0, (size_t)N_PTS * 4 * 4, stream);
    spconv96x4<<<dim3(EB, CONV_MAPS), 256, 0, stream>>>(qf1, Wq2, kmaps, qf4);
    hipMemsetAsync(stats, 0, 1024, stream);
    colstats<<<1024, 256, 0, stream>>>(qf4, stats, (long long)N_PTS * 4, 4);
    bn_apply<<<EL4, 256, 0, stream>>>(qf4, qf4, stats, gq2, bq2, nullptr, nullptr, N_PTS, 4);

    // vf_pre = spconv(x, Wv)
    hipMemsetAsync(vf, 0, (size_t)N_PTS * 96 * 4, stream);
    spconv96x96_wmma<<<dim3(NT, CONV_MAPS), 256, 0, stream>>>(x, Wv, kmaps, vf);

    // pe = relu(bn(relu(bn(coords@Wp1)) @ Wp2))
    pe_h1<<<EL96, 256, 0, stream>>>(coords, Wp1, h1, N_PTS);
    hipMemsetAsync(stats, 0, 1024, stream);
    colstats<<<1024, 256, 0, stream>>>(h1, stats, (long long)N_PTS * 96, 96);
    bn_apply<<<EL96, 256, 0, stream>>>(h1, h1, stats, gp1, bp1, nullptr, nullptr, N_PTS, 96);
    pe_h2<<<(N_PTS + 255) / 256, 256, 0, stream>>>(h1, Wp2, pe4, N_PTS);
    hipMemsetAsync(stats, 0, 1024, stream);
    colstats<<<1024, 256, 0, stream>>>(pe4, stats, (long long)N_PTS * 4, 4);
    bn_apply<<<EL4, 256, 0, stream>>>(pe4, pe4, stats, gp2, bp2, nullptr, nullptr, N_PTS, 4);

    // v_f = relu(bn(vf_pre)) + repeat(pe, 24)
    hipMemsetAsync(stats, 0, 1024, stream);
    colstats<<<1024, 256, 0, stream>>>(vf, stats, (long long)N_PTS * 96, 96);
    bn_apply<<<EL96, 256, 0, stream>>>(vf, vf, stats, gv, bv, pe4, nullptr, N_PTS, 96);

    // inverse maps, logits, softmax
    hipMemsetAsync(inv, 0xFF, (size_t)K_MAPS * N_PTS * 4, stream);
    build_inv<<<(int)(((long long)K_MAPS * M_EDGES + 255) / 256), 256, 0, stream>>>(kmaps, inv);
    attn_logits<<<(int)(((long long)K_MAPS * N_PTS + 7) / 8), 256, 0, stream>>>(
        x, qf4, inv, W1, b1, W2, b2, logits, masks);
    attn_softmax<<<EL4, 256, 0, stream>>>(logits);

    // out = relu(bn(sum_k ...)) + x
    out_accum<<<(N_PTS + 7) / 8, 256, 0, stream>>>(vf, logits, masks, inv, outp);
    hipMemsetAsync(stats, 0, 1024, stream);
    colstats<<<1024, 256, 0, stream>>>(outp, stats, (long long)N_PTS * 96, 96);
    bn_apply<<<EL96, 256, 0, stream>>>(outp, (float*)d_out, stats, g_out, b_out,
                                       nullptr, x, N_PTS, 96);
}